// NetSAGE_32504312496300
// MI455X (gfx1250) — compile-verified
//
#include <hip/hip_runtime.h>

#define N_NODES 100000
#define N_EDGES 1600000
#define NFEAT 128
#define NHID 128
#define NCLASS 8
#define ASTRIDE 132   // padded LDS row stride (floats): bank = (4*row + k) % 64

typedef float v2f __attribute__((ext_vector_type(2)));
typedef float v8f __attribute__((ext_vector_type(8)));

// ---------------------------------------------------------------------------
// Zero a float buffer (float4 granularity). n4 = number of float4 elements.
// ---------------------------------------------------------------------------
__global__ void sage_zero_kernel(float4* __restrict__ p, int n4) {
    int i = blockIdx.x * blockDim.x + threadIdx.x;
    if (i < n4) p[i] = make_float4(0.f, 0.f, 0.f, 0.f);
}

// ---------------------------------------------------------------------------
// Transpose a 128x128 f32 matrix: out[c][k] = in[k][c]. One-time, 64 KB.
// ---------------------------------------------------------------------------
__global__ void sage_transpose128_kernel(const float* __restrict__ in,
                                         float* __restrict__ out) {
    int i = blockIdx.x * blockDim.x + threadIdx.x;   // 0..16383
    int c = i >> 7;
    int k = i & 127;
    out[c * NHID + k] = in[k * NHID + c];
}

// ---------------------------------------------------------------------------
// Edge scatter-add: agg[dst] += feat[src]. One wave32 per edge; each lane
// loads one float4 (128 floats / 32 lanes = 4 floats per lane) and issues
// four global_atomic_add_f32. Feature matrix fits in L2 (51.2 MB << 192 MB),
// so gathers and atomics are L2-resident.
// ---------------------------------------------------------------------------
__global__ void sage_scatter_add_kernel(const float* __restrict__ feat,
                                        const int* __restrict__ esrc,
                                        const int* __restrict__ edst,
                                        float* __restrict__ agg) {
    unsigned gid  = blockIdx.x * blockDim.x + threadIdx.x;
    unsigned edge = gid >> 5;
    unsigned lane = gid & 31u;
    if (edge >= (unsigned)N_EDGES) return;
    int s = esrc[edge];
    int d = edst[edge];
    const float4 v = reinterpret_cast<const float4*>(feat + (size_t)s * NFEAT)[lane];
    float* dst = agg + (size_t)d * NFEAT + lane * 4;
    atomicAdd(dst + 0, v.x);
    atomicAdd(dst + 1, v.y);
    atomicAdd(dst + 2, v.z);
    atomicAdd(dst + 3, v.w);
}

// ---------------------------------------------------------------------------
// Layer-1 fused GEMM: h = relu(agg @ Wn + x @ Ws + b), fp32 via
// V_WMMA_F32_16X16X4_F32.
//
// One 256-thread block = 8 waves = one 16-row tile x all 8 column tiles.
// The 16x128 A tiles of BOTH input matrices (agg, x) are staged once per
// block into LDS with GLOBAL_LOAD_ASYNC_TO_LDS_B128 (ASYNCcnt); WMMA
// A-fragments then come from ds_load (fused to ds_load_2addr_b64) with a
// padded row stride (132 floats) so lanes 0-15 hit distinct banks.
// B fragments are single global_load_b64 from the PRE-TRANSPOSED weights
// (WT[col][ka..ka+1]), halving hot-loop VMEM instruction count vs strided
// per-element b32 loads of the row-major weights.
//
// Fragment layouts (ISA 7.12.2):
//   A 16x4 f32 : lane L -> row M = L%16, K = 2*(L/16) + {0,1}
//   B 4x16 f32 : lane L -> col N = L%16, K = 2*(L/16) + {0,1}
//   C/D 16x16  : lane L, VGPR r -> M = r + 8*(L/16), N = L%16
// ---------------------------------------------------------------------------
__global__ void sage_gemm1_kernel(const float* __restrict__ agg,
                                  const float* __restrict__ x,
                                  const float* __restrict__ WnT,
                                  const float* __restrict__ WsT,
                                  const float* __restrict__ bias,
                                  float* __restrict__ h) {
    __shared__ float tile[2 * 16 * ASTRIDE];    // [mat][row][k], 16.9 KB

    const int t       = threadIdx.x;
    const int rowTile = blockIdx.x;             // 0..6249 (100000 = 6250*16)
    const unsigned rowBaseByte = (unsigned)rowTile * 16u * NFEAT * 4u;

    // ---- async stage: 2 matrices x 16 rows x 512 B = 1024 b128 chunks ------
    {
        const unsigned ldsBase = (unsigned)(unsigned long long)(uintptr_t)tile;
        #pragma unroll
        for (int j = 0; j < 4; ++j) {
            const int chunk = t + j * 256;      // 0..1023
            const int mat   = chunk >> 9;       // constant-folded per j
            const int rem   = chunk & 511;
            const int row   = rem >> 5;         // 0..15
            const int c16   = rem & 31;         // 16B chunk within the row
            const unsigned goff = rowBaseByte + (unsigned)(row * NFEAT + c16 * 4) * 4u;
            const unsigned loff = ldsBase +
                (unsigned)((mat * 16 + row) * ASTRIDE + c16 * 4) * 4u;
            const float* gbase = mat ? x : agg;
            asm volatile("global_load_async_to_lds_b128 %0, %1, %2"
                         :: "v"(loff), "v"(goff), "s"(gbase) : "memory");
        }
        asm volatile("s_wait_asynccnt 0x0" ::: "memory");
    }
    __syncthreads();

    const int wave  = t >> 5;                   // 0..7 -> column tile
    const int lane  = t & 31;
    const int lhalf = lane >> 4;                // 0 or 1
    const int l16   = lane & 15;
    const int col   = wave * 16 + l16;          // B-fragment column
    const int khalf = lhalf * 2;                // K offset within K=4 step

    const int aRowOff = l16 * ASTRIDE;          // agg tile row (LDS)
    const int xRowOff = (16 + l16) * ASTRIDE;   // x   tile row (LDS)
    const float* WnCol = WnT + (size_t)col * NHID;   // column of original Wn
    const float* WsCol = WsT + (size_t)col * NHID;

    v8f c = {};
    #pragma unroll 4
    for (int k0 = 0; k0 < NFEAT; k0 += 4) {
        const int ka = k0 + khalf;              // even -> 8B-aligned b64 loads
        const v2f aA = *reinterpret_cast<const v2f*>(&tile[aRowOff + ka]);
        const v2f aX = *reinterpret_cast<const v2f*>(&tile[xRowOff + ka]);
        const v2f bN = *reinterpret_cast<const v2f*>(WnCol + ka);
        const v2f bS = *reinterpret_cast<const v2f*>(WsCol + ka);
        // 8 args: (neg_a, A, neg_b, B, c_mod, C, reuse_a, reuse_b)
        c = __builtin_amdgcn_wmma_f32_16x16x4_f32(false, aA, false, bN,
                                                  (short)0, c, false, false);
        c = __builtin_amdgcn_wmma_f32_16x16x4_f32(false, aX, false, bS,
                                                  (short)0, c, false, false);
    }

    const float bval  = bias[col];
    const int   rbase = rowTile * 16 + 8 * lhalf;   // C-layout row base
    #pragma unroll
    for (int r = 0; r < 8; ++r) {
        float v = c[r] + bval;
        h[(size_t)(rbase + r) * NHID + col] = v > 0.f ? v : 0.f;
    }
}

// ---------------------------------------------------------------------------
// Layer-2 (NCLASS=8, negligible FLOPs) + log_softmax, fused. One wave32 per
// node row: lane owns class (lane&7) and K-chunk (lane>>3); reduce chunks via
// shfl_xor(8,16), then log-softmax across the 8 classes via shfl_xor(1,2,4).
// ---------------------------------------------------------------------------
__global__ void sage_layer2_kernel(const float* __restrict__ agg,
                                   const float* __restrict__ h,
                                   const float* __restrict__ Wn,
                                   const float* __restrict__ Ws,
                                   const float* __restrict__ bias,
                                   float* __restrict__ out) {
    const int row  = blockIdx.x * (blockDim.x >> 5) + (threadIdx.x >> 5);
    const int lane = threadIdx.x & 31;
    const int cls  = lane & 7;
    const int kcnk = lane >> 3;                  // 0..3, 32 K each
    if (row >= N_NODES) return;

    const float* aRow = agg + (size_t)row * NHID;
    const float* hRow = h   + (size_t)row * NHID;
    float acc = 0.f;
    #pragma unroll 8
    for (int k = kcnk * 32; k < kcnk * 32 + 32; ++k) {
        acc = fmaf(aRow[k], Wn[k * NCLASS + cls], acc);
        acc = fmaf(hRow[k], Ws[k * NCLASS + cls], acc);
    }
    // sum the 4 K-chunks (lanes cls, cls+8, cls+16, cls+24)
    acc += __shfl_xor(acc, 8, 32);
    acc += __shfl_xor(acc, 16, 32);
    const float v = acc + bias[cls];

    // log_softmax across the 8 classes (lanes close under xor {1,2,4})
    float m = v;
    m = fmaxf(m, __shfl_xor(m, 1, 32));
    m = fmaxf(m, __shfl_xor(m, 2, 32));
    m = fmaxf(m, __shfl_xor(m, 4, 32));
    float e = __expf(v - m);
    float s = e;
    s += __shfl_xor(s, 1, 32);
    s += __shfl_xor(s, 2, 32);
    s += __shfl_xor(s, 4, 32);
    if (lane < 8)
        out[(size_t)row * NCLASS + cls] = (v - m) - __logf(s);
}

// ---------------------------------------------------------------------------
// Launch pipeline (all on `stream`, graph-capture safe):
//   transpose(Wn1), transpose(Ws1)
//   zero(agg) -> scatter(x->agg) -> gemm1(agg,x->h)
//   -> zero(agg) -> scatter(h->agg) -> layer2(agg,h->out)
// Workspace: agg (51.2 MB) + h (51.2 MB) + WnT + WsT (128 KB).
// ---------------------------------------------------------------------------
extern "C" void kernel_launch(void* const* d_in, const int* in_sizes, int n_in,
                              void* d_out, int out_size, void* d_ws, size_t ws_size,
                              hipStream_t stream) {
    const float* x    = (const float*)d_in[0];
    const int*   esrc = (const int*)d_in[1];
    const int*   edst = (const int*)d_in[2];
    const float* Wn1  = (const float*)d_in[3];
    const float* Ws1  = (const float*)d_in[4];
    const float* b1   = (const float*)d_in[5];
    const float* Wn2  = (const float*)d_in[6];
    const float* Ws2  = (const float*)d_in[7];
    const float* b2   = (const float*)d_in[8];
    float*       out  = (float*)d_out;

    float* agg = (float*)d_ws;
    float* h   = agg + (size_t)N_NODES * NHID;
    float* WnT = h   + (size_t)N_NODES * NHID;
    float* WsT = WnT + (size_t)NFEAT * NHID;

    const int n4          = N_NODES * NFEAT / 4;            // 3,200,000 float4
    const int zeroBlocks  = (n4 + 255) / 256;
    const int scatBlocks  = (int)(((long long)N_EDGES * 32) / 256);  // 200,000
    const int gemm1Blocks = N_NODES / 16;                   // 6,250
    const int l2Blocks    = (N_NODES + 7) / 8;              // 12,500 (8 waves/blk)
    const int trBlocks    = (NFEAT * NHID) / 256;           // 64

    sage_transpose128_kernel<<<trBlocks, 256, 0, stream>>>(Wn1, WnT);
    sage_transpose128_kernel<<<trBlocks, 256, 0, stream>>>(Ws1, WsT);
    sage_zero_kernel<<<zeroBlocks, 256, 0, stream>>>((float4*)agg, n4);
    sage_scatter_add_kernel<<<scatBlocks, 256, 0, stream>>>(x, esrc, edst, agg);
    sage_gemm1_kernel<<<gemm1Blocks, 256, 0, stream>>>(agg, x, WnT, WsT, b1, h);
    sage_zero_kernel<<<zeroBlocks, 256, 0, stream>>>((float4*)agg, n4);
    sage_scatter_add_kernel<<<scatBlocks, 256, 0, stream>>>(h, esrc, edst, agg);
    sage_layer2_kernel<<<l2Blocks, 256, 0, stream>>>(agg, h, Wn2, Ws2, b2, out);
}